// SamplePredictor_2104533975212
// MI455X (gfx1250) — compile-verified
//
#include <hip/hip_runtime.h>
#include <hip/hip_bf16.h>

#define HID 64

typedef __attribute__((ext_vector_type(2))) float v2f;
typedef __attribute__((ext_vector_type(8))) float v8f;

// ---------------------------------------------------------------- degree
__global__ void __launch_bounds__(256)
deg_kernel(const int* __restrict__ dst, float* __restrict__ deg, int E) {
    int t = blockIdx.x * blockDim.x + threadIdx.x;
    if (t < E) unsafeAtomicAdd(&deg[dst[t]], 1.0f);
}

// dinv[i] = 1/sqrt(deg[i] + 1)   (+1 = self loop, always > 0)
__global__ void __launch_bounds__(256)
dinv_kernel(const float* __restrict__ deg, float* __restrict__ dinv, int N) {
    int t = blockIdx.x * blockDim.x + threadIdx.x;
    if (t < N) dinv[t] = rsqrtf(deg[t] + 1.0f);
}

// ---------------------------------------------------------------- x @ W1 (K=5)
__global__ void __launch_bounds__(256)
xw1_kernel(const float* __restrict__ x, const float* __restrict__ w1,
           float* __restrict__ out, int N) {
    long long t = (long long)blockIdx.x * blockDim.x + threadIdx.x;
    if (t >= (long long)N * HID) return;
    int n = (int)(t >> 6);
    int f = (int)(t & 63);
    const float* xr = x + (long long)n * 5;
    float s = 0.f;
#pragma unroll
    for (int i = 0; i < 5; ++i) s += xr[i] * w1[i * HID + f];
    out[t] = s;
}

// ---------------------------------------------------------------- edge scatter
// 16 threads per (virtual) edge; edges [0,E) real, [E,E+N) self loops.
__global__ void __launch_bounds__(256)
scatter_kernel(const float* __restrict__ h, const int* __restrict__ src,
               const int* __restrict__ dst, const float* __restrict__ dinv,
               float* __restrict__ acc, long long totalE, int E) {
    long long t = (long long)blockIdx.x * blockDim.x + threadIdx.x;
    long long e = t >> 4;
    if (e >= totalE) return;
    int f4 = (int)(t & 15) * 4;
    int s, d;
    if (e < (long long)E) { s = src[e]; d = dst[e]; }
    else                  { s = d = (int)(e - E); }
    float nrm = dinv[s] * dinv[d];
    const float4 hv = *(const float4*)(h + (long long)s * HID + f4);
    float* ap = acc + (long long)d * HID + f4;
    unsafeAtomicAdd(ap + 0, hv.x * nrm);
    unsafeAtomicAdd(ap + 1, hv.y * nrm);
    unsafeAtomicAdd(ap + 2, hv.z * nrm);
    unsafeAtomicAdd(ap + 3, hv.w * nrm);
}

// ---------------------------------------------------------------- bias + relu (in place)
__global__ void __launch_bounds__(256)
bias_relu_kernel(float* __restrict__ a, const float* __restrict__ b, long long total) {
    long long t = (long long)blockIdx.x * blockDim.x + threadIdx.x;
    if (t >= total) return;
    float v = a[t] + b[(int)(t & 63)];
    a[t] = v > 0.f ? v : 0.f;
}

// ---------------------------------------------------------------- h1 @ W2 via V_WMMA_F32_16X16X4_F32
// Block = 4 waves; wave w computes D[m0:m0+16, 16w:16w+16], K=64 in 16 steps.
// A 16x4 layout: lanes 0-15 hold (M=lane, K=kh..kh+1) with kh=0; lanes 16-31 kh=2.
// B 4x16 layout mirrors A over N.  C/D: VGPR v = row (v | lane>=16 ? v+8), col = lane&15.
__global__ void __launch_bounds__(128)
gemm_wmma_kernel(const float* __restrict__ A, const float* __restrict__ B,
                 float* __restrict__ D, int N) {
    int m0   = blockIdx.x * 16;
    int n0   = (threadIdx.x >> 5) * 16;
    int lane = threadIdx.x & 31;
    int row  = lane & 15;
    int kh   = (lane >> 4) * 2;

    int ar = m0 + row; if (ar >= N) ar = N - 1;      // clamp (N%16==0 normally)
    const float* arow = A + (long long)ar * HID;

    v8f c = {};
#pragma unroll
    for (int k = 0; k < HID; k += 4) {
        v2f av, bv;
        av.x = arow[k + kh];
        av.y = arow[k + kh + 1];
        bv.x = B[(k + kh)     * HID + n0 + row];
        bv.y = B[(k + kh + 1) * HID + n0 + row];
        c = __builtin_amdgcn_wmma_f32_16x16x4_f32(false, av, false, bv,
                                                  (short)0, c, false, false);
    }
    int mb  = m0 + ((lane >> 4) ? 8 : 0);
    int col = n0 + row;
#pragma unroll
    for (int v = 0; v < 8; ++v)
        if (mb + v < N) D[(long long)(mb + v) * HID + col] = c[v];
}

// ---------------------------------------------------------------- relu(acc2+b2) -> mean pool (sum)
__global__ void __launch_bounds__(256)
pool_kernel(const float* __restrict__ acc, const float* __restrict__ bias,
            float* __restrict__ poolsum, int N) {
    __shared__ float sdata[256];
    int f = threadIdx.x & 63;
    int r = threadIdx.x >> 6;
    float bf = bias[f];
    float local = 0.f;
    for (int n = blockIdx.x * 4 + r; n < N; n += gridDim.x * 4) {
        float v = acc[(long long)n * HID + f] + bf;
        local += v > 0.f ? v : 0.f;
    }
    sdata[threadIdx.x] = local;
    __syncthreads();
    if (r == 0) {
        float s = sdata[f] + sdata[64 + f] + sdata[128 + f] + sdata[192 + f];
        unsafeAtomicAdd(&poolsum[f], s);
    }
}

// ---------------------------------------------------------------- global features over x
// gsum = [sum x2, sum x3, sum x4, sum(x2==1), sum x0*m, sum x1*m]
__global__ void __launch_bounds__(256)
gfeat_kernel(const float* __restrict__ x, float* __restrict__ gsum, int N) {
    float l[6] = {0.f, 0.f, 0.f, 0.f, 0.f, 0.f};
    for (int n = blockIdx.x * blockDim.x + threadIdx.x; n < N;
         n += gridDim.x * blockDim.x) {
        const float* xr = x + (long long)n * 5;
        float x0 = xr[0], x1 = xr[1], x2 = xr[2], x3 = xr[3], x4 = xr[4];
        float m = (x2 == 1.0f) ? 1.0f : 0.0f;
        l[0] += x2; l[1] += x3; l[2] += x4;
        l[3] += m;  l[4] += x0 * m; l[5] += x1 * m;
    }
    __shared__ float sdata[256];
#pragma unroll
    for (int q = 0; q < 6; ++q) {
        sdata[threadIdx.x] = l[q];
        __syncthreads();
        for (int s = 128; s > 0; s >>= 1) {
            if (threadIdx.x < (unsigned)s) sdata[threadIdx.x] += sdata[threadIdx.x + s];
            __syncthreads();
        }
        if (threadIdx.x == 0) unsafeAtomicAdd(&gsum[q], sdata[0]);
        __syncthreads();
    }
}

// ---------------------------------------------------------------- MLP head (one wave)
__global__ void __launch_bounds__(32)
mlp_kernel(const float* __restrict__ poolsum, const float* __restrict__ gsum,
           const float* __restrict__ l1w, const float* __restrict__ l1b,
           const float* __restrict__ l2w, const float* __restrict__ l2b,
           float* __restrict__ out, int N) {
    __shared__ float hid[32];
    int j = threadIdx.x;
    float invN = 1.0f / (float)N;
    float acc = l1b[j];
    for (int i = 0; i < HID; ++i)
        acc += (poolsum[i] * invN) * l1w[i * 32 + j];
    float s2 = gsum[0], s3 = gsum[1], s4 = gsum[2];
    float cnt = gsum[3], s0m = gsum[4], s1m = gsum[5];
    float safe = cnt > 1.0f ? cnt : 1.0f;
    float avgl = cnt > 0.f ? s0m / safe : 0.f;
    float avgm = cnt > 0.f ? s1m / safe : 0.f;
    float gf[6] = {s2, s3, s4, s3 + s4, avgl, avgm};
#pragma unroll
    for (int i = 0; i < 6; ++i)
        acc += gf[i] * l1w[(HID + i) * 32 + j];
    hid[j] = acc > 0.f ? acc : 0.f;
    __syncthreads();
    if (j < 2) {
        float o = l2b[j];
        for (int q = 0; q < 32; ++q) o += hid[q] * l2w[q * 2 + j];
        float sg = 1.0f / (1.0f + __expf(-o));
        out[j] = 2.0f + sg * 3.0f;
    }
}

// ----------------------------------------------------------------
extern "C" void kernel_launch(void* const* d_in, const int* in_sizes, int n_in,
                              void* d_out, int out_size, void* d_ws, size_t ws_size,
                              hipStream_t stream) {
    const float* x   = (const float*)d_in[0];
    const int*   ei  = (const int*)d_in[1];
    const float* w1  = (const float*)d_in[2];
    const float* b1  = (const float*)d_in[3];
    const float* w2  = (const float*)d_in[4];
    const float* b2  = (const float*)d_in[5];
    const float* l1w = (const float*)d_in[6];
    const float* l1b = (const float*)d_in[7];
    const float* l2w = (const float*)d_in[8];
    const float* l2b = (const float*)d_in[9];

    const int N = in_sizes[0] / 5;
    const int E = in_sizes[1] / 2;
    const int* src = ei;
    const int* dst = ei + E;

    float* ws      = (float*)d_ws;
    float* deg     = ws;                              // N
    float* dinv    = ws + N;                          // N
    float* bufA    = ws + 2 * (size_t)N;              // N*64
    float* bufB    = bufA + (size_t)N * HID;          // N*64
    float* poolsum = bufB + (size_t)N * HID;          // 64
    float* gsum    = poolsum + 64;                    // 8

    const long long totalE = (long long)E + N;
    const long long nh     = (long long)N * HID;

    // zero accumulators (capturable)
    hipMemsetAsync(deg, 0, sizeof(float) * (size_t)N, stream);
    hipMemsetAsync(bufB, 0, sizeof(float) * (size_t)nh, stream);
    hipMemsetAsync(poolsum, 0, sizeof(float) * 72, stream);

    deg_kernel <<<(E + 255) / 256, 256, 0, stream>>>(dst, deg, E);
    dinv_kernel<<<(N + 255) / 256, 256, 0, stream>>>(deg, dinv, N);

    // conv1
    xw1_kernel<<<(unsigned)((nh + 255) / 256), 256, 0, stream>>>(x, w1, bufA, N);
    long long st1 = totalE * 16;
    scatter_kernel<<<(unsigned)((st1 + 255) / 256), 256, 0, stream>>>(
        bufA, src, dst, dinv, bufB, totalE, E);
    bias_relu_kernel<<<(unsigned)((nh + 255) / 256), 256, 0, stream>>>(bufB, b1, nh);

    // conv2: WMMA GEMM h1 @ W2  -> bufA
    gemm_wmma_kernel<<<(N + 15) / 16, 128, 0, stream>>>(bufB, w2, bufA, N);
    hipMemsetAsync(bufB, 0, sizeof(float) * (size_t)nh, stream);
    scatter_kernel<<<(unsigned)((st1 + 255) / 256), 256, 0, stream>>>(
        bufA, src, dst, dinv, bufB, totalE, E);

    // mean pool (fused bias+relu) + global features + head
    pool_kernel <<<256, 256, 0, stream>>>(bufB, b2, poolsum, N);
    gfeat_kernel<<<256, 256, 0, stream>>>(x, gsum, N);
    mlp_kernel  <<<1, 32, 0, stream>>>(poolsum, gsum, l1w, l1b, l2w, l2b,
                                       (float*)d_out, N);
}